// HMM_NER_23287312679365
// MI455X (gfx1250) — compile-verified
//
#include <hip/hip_runtime.h>

// ---------------------------------------------------------------------------
// Viterbi (max-product HMM) persistent kernel for MI455X / gfx1250.
//
// 64 persistent WGs x 256 threads; each WG owns 32 columns of the 2048x2048
// transition matrix. Its 2048x32 column block (256 KB) is DMA'd ONCE into the
// 320 KB WGP LDS via global_load_async_to_lds_b128 (ASYNCcnt path) and reused
// for all 4096 steps. Per step: ~8 KB global traffic + ONE fused
// flag-publish sync (release tag store + relaxed polls + acquire fence) that
// doubles as the cross-WG max/argmax reduction. Max-product is a tropical
// semiring, so WMMA is inapplicable; the CDNA5 paths used are async-LDS DMA,
// wave32 shuffles, LDS, and the split wait counters.
// ---------------------------------------------------------------------------

#define NT        2048          // number of tags
#define VOCAB     32000
#define NWG       64            // persistent workgroups (all co-resident)
#define COLS      32            // trans columns per WG  (NT / NWG)
#define NTHREADS  256
#define SLICES    (NTHREADS / COLS)   // 8 j-slices
#define JPS       (NT / SLICES)       // 256 j-values per slice
#define SMEM_FLOATS (NT * COLS + NT + NTHREADS)
#define SMEM_BYTES  (SMEM_FLOATS * 4) // 271360 B < 320 KB WGP LDS

// -------- CDNA5 async global->LDS DMA (ASYNCcnt path) ----------------------
__device__ __forceinline__ void async_load_b128(const void* gptr, void* lptr) {
  // Low 32 bits of a generic pointer into LDS are the LDS byte offset.
  unsigned lds_off = (unsigned)(unsigned long long)lptr;
  asm volatile("global_load_async_to_lds_b128 %0, %1, off"
               :: "v"(lds_off), "v"(gptr) : "memory");
}

__device__ __forceinline__ void wait_async_zero() {
#if __has_builtin(__builtin_amdgcn_s_wait_asynccnt)
  __builtin_amdgcn_s_wait_asynccnt(0);
#else
  asm volatile("s_wait_asynccnt 0" ::: "memory");
#endif
}

__global__ void viterbi_init_ws(unsigned* pairsTag) {
  // zero the 2*NWG publication tags (fresh every launch -> graph-replay safe)
  if (threadIdx.x < 2 * NWG) pairsTag[threadIdx.x] = 0u;
}

// ---------------------------------------------------------------------------
__global__ __launch_bounds__(NTHREADS)
void viterbi_kernel(const float* __restrict__ emiss,   // [NT, VOCAB]
                    const float* __restrict__ trans,   // [NT, NT]
                    const int*   __restrict__ seq,     // [T]
                    const int*   __restrict__ o_tag_p, // [1]
                    int T,
                    int*      __restrict__ out,        // [T] int32
                    unsigned* pairsTag,                // [2][NWG] step tags
                    float*    pairsVal,                // [2][NWG]
                    int*      pairsIdx,                // [2][NWG]
                    float*    prob) {                  // [2][NT] ping-pong
  extern __shared__ float smem[];
  float* transLDS = smem;                 // [NT][COLS]  256 KB, loaded once
  float* prevLDS  = smem + NT * COLS;     // [NT]        8 KB
  float* redLDS   = prevLDS + NT;         // [NTHREADS]  1 KB

  const int wg    = blockIdx.x;
  const int tid   = threadIdx.x;
  const int col   = tid & (COLS - 1);     // 0..31 (exactly one wave32 wide)
  const int slice = tid >> 5;             // 0..7
  const int o_tag = o_tag_p[0];

  // ---- one-time DMA of this WG's 2048x32 trans column block into LDS ----
  for (int r = tid; r < NT; r += NTHREADS) {
    const float* g = trans + (size_t)r * NT + wg * COLS;   // 128B row slice
    float*       l = transLDS + r * COLS;
#pragma unroll
    for (int q = 0; q < COLS; q += 4)
      async_load_b128(g + q, l + q);
  }
  wait_async_zero();
  __syncthreads();

  bool fallback = false;

  for (int t = 0; t < T; ++t) {
    const int p = t & 1;                  // parity written this step
    const int q = p ^ 1;                  // parity read this step (t-1)
    const int word = seq[t];
    float cur;                            // valid in wave 0 (tid < COLS)

    if (t == 0) {
      // prob0[i] = emiss[i, seq[0]] * trans[0, i]   (no fallback at t=0)
      if (tid < COLS) {
        const int gi = wg * COLS + col;
        cur = emiss[(size_t)gi * VOCAB + word] * transLDS[col]; // trans[0][gi]
      }
    } else {
      // Stage prev (parity q) into LDS, applying the one-hot fallback.
      const float4* psrc = (const float4*)(prob + q * NT);
      for (int k = tid; k < NT / 4; k += NTHREADS) {
        float4 v = psrc[k];
        if (fallback) {
          const int b = 4 * k;
          v.x = (b + 0 == o_tag) ? 1.0f : 0.0f;
          v.y = (b + 1 == o_tag) ? 1.0f : 0.0f;
          v.z = (b + 2 == o_tag) ? 1.0f : 0.0f;
          v.w = (b + 3 == o_tag) ? 1.0f : 0.0f;
        }
        *(float4*)&prevLDS[4 * k] = v;
      }
      __syncthreads();

      // m = max over this slice's 256 j of prev[j] * trans[j, col]
      float m = 0.0f;                     // probabilities are >= 0
      const int jbase = slice * JPS;
#pragma unroll 4
      for (int j = 0; j < JPS; j += 4) {
        float4 pv = *(const float4*)&prevLDS[jbase + j];     // ds_load_b128
        const float* trow = &transLDS[(jbase + j) * COLS + col];
        m = fmaxf(m, pv.x * trow[0 * COLS]);                 // conflict-free
        m = fmaxf(m, pv.y * trow[1 * COLS]);
        m = fmaxf(m, pv.z * trow[2 * COLS]);
        m = fmaxf(m, pv.w * trow[3 * COLS]);
      }
      redLDS[tid] = m;
      __syncthreads();

      if (tid < COLS) {
        float mm = redLDS[col];
#pragma unroll
        for (int s = 1; s < SLICES; ++s) mm = fmaxf(mm, redLDS[s * COLS + col]);
        const int gi = wg * COLS + col;
        cur = mm * emiss[(size_t)gi * VOCAB + word];
      }
    }

    // ---- publish: prob chunk + (max, argmax) pair + release tag ----------
    if (tid < COLS) {
      const int gi = wg * COLS + col;
      prob[p * NT + gi] = cur;

      float bv = cur; int bi = gi;
#pragma unroll
      for (int off = 16; off > 0; off >>= 1) {               // wave32 argmax
        float ov = __shfl_xor(bv, off, 32);
        int   oi = __shfl_xor(bi, off, 32);
        if (ov > bv || (ov == bv && oi < bi)) { bv = ov; bi = oi; }
      }
      if (col == 0) {
        pairsVal[p * NWG + wg] = bv;
        pairsIdx[p * NWG + wg] = bi;
        // Release: waits on this wave's outstanding stores (prob + pair),
        // then makes the tag visible at agent scope.
        __hip_atomic_store(&pairsTag[p * NWG + wg], (unsigned)t + 1u,
                           __ATOMIC_RELEASE, __HIP_MEMORY_SCOPE_AGENT);
      }
    }

    // ---- fused sync + cross-WG reduction ---------------------------------
    const unsigned want = (unsigned)t + 1u;
    if (tid < NWG) {
      // Relaxed agent-scope load: scope DEV > WGP$ scope, so it always goes
      // to L2 (coherent; no stale-spin). Tags are monotonic, so '>=' is safe
      // even if a fast WG already republished this parity two steps later.
      while (__hip_atomic_load(&pairsTag[p * NWG + tid],
                               __ATOMIC_RELAXED, __HIP_MEMORY_SCOPE_AGENT) < want) {
        __builtin_amdgcn_s_sleep(1);
      }
    }
    __syncthreads();
    __threadfence();   // acquire: invalidate WGP$ before reading pairs/prob

    // Every WG derives the same global max / argmax / fallback flag.
    float gv = -1.0f; int gidx = 0;
    for (int w = 0; w < NWG; ++w) {
      float v = pairsVal[p * NWG + w];
      if (v > gv) { gv = v; gidx = pairsIdx[p * NWG + w]; }
    }
    if (t == 0) {
      fallback = false;                       // reference applies none at t=0
      if (wg == 0 && tid == 0) out[0] = gidx;
    } else {
      fallback = (gv == 0.0f);
      if (wg == 0 && tid == 0) out[t] = fallback ? o_tag : gidx;
    }
  }
}

// ---------------------------------------------------------------------------
extern "C" void kernel_launch(void* const* d_in, const int* in_sizes, int n_in,
                              void* d_out, int out_size, void* d_ws, size_t ws_size,
                              hipStream_t stream) {
  const float* emiss = (const float*)d_in[0];   // [2048, 32000] f32
  const float* trans = (const float*)d_in[1];   // [2048, 2048]  f32
  const int*   seq   = (const int*)d_in[2];     // [4096] i32
  const int*   o_tag = (const int*)d_in[3];     // [1] i32
  int*         out   = (int*)d_out;             // [4096] i32
  const int T = in_sizes[2];

  // workspace carve (64B-aligned regions)
  unsigned char* ws = (unsigned char*)d_ws;
  unsigned* pairsTag = (unsigned*)(ws + 0);     // 2*NWG u32  = 512 B
  float*    pairsVal = (float*)(ws + 512);      // 2*NWG f32  = 512 B
  int*      pairsIdx = (int*)(ws + 1024);       // 2*NWG i32  = 512 B
  float*    prob     = (float*)(ws + 1536);     // 2*NT  f32  = 16 KB

  // 271,360 B dynamic LDS (< 320 KB WGP LDS on CDNA5)
  (void)hipFuncSetAttribute((const void*)viterbi_kernel,
                            hipFuncAttributeMaxDynamicSharedMemorySize,
                            SMEM_BYTES);

  viterbi_init_ws<<<1, 2 * NWG, 0, stream>>>(pairsTag);
  viterbi_kernel<<<NWG, NTHREADS, SMEM_BYTES, stream>>>(
      emiss, trans, seq, o_tag, T, out, pairsTag, pairsVal, pairsIdx, prob);
}